// GCN_28638841930196
// MI455X (gfx1250) — compile-verified
//
#include <hip/hip_runtime.h>

#define NN 100000
#define NE 1600000

typedef __attribute__((ext_vector_type(2))) float v2f;
typedef __attribute__((ext_vector_type(8))) float v8f;

// ---------------- utility kernels ----------------

__global__ void zero_f32(float* __restrict__ p, size_t n) {
    size_t i = (size_t)blockIdx.x * blockDim.x + threadIdx.x;
    size_t stride = (size_t)gridDim.x * blockDim.x;
    for (; i < n; i += stride) p[i] = 0.0f;
}

__global__ void degree_kernel(const int* __restrict__ dst, float* __restrict__ deg, int nE) {
    int e = blockIdx.x * blockDim.x + threadIdx.x;
    if (e < nE) atomicAdd(&deg[dst[e]], 1.0f);
}

// in-place: deg -> 1/sqrt(deg + 1)   (+1 = self loop; always > 0)
__global__ void rsqrt_kernel(float* __restrict__ d, int n) {
    int i = blockIdx.x * blockDim.x + threadIdx.x;
    if (i < n) d[i] = rsqrtf(d[i] + 1.0f);
}

// ---------------- WMMA tall-skinny GEMM ----------------
// out[N,32] = concat(inA[N,KA], inB[N,KB]) @ W[(KA+KB),32]  (+bias[col]) (*scale[row])
// One wave32 computes a 16x32 tile via V_WMMA_F32_16X16X4_F32.
// A frag (16x4 f32): lanes 0-15 hold M=lane, K={k,k+1}; lanes 16-31 hold M=lane-16, K={k+2,k+3}
// B frag (4x16 f32): lanes 0-15 hold K rows {k,k+1}, N=lane&15; lanes 16-31 hold K rows {k+2,k+3}
// C/D (16x16 f32):   vgpr v: lanes 0-15 -> M=v, lanes 16-31 -> M=v+8; N = lane&15
template <int KA, int KB, bool BIAS, bool SCALE>
__global__ __launch_bounds__(256)
void gemm_tiles(const float* __restrict__ inA, int lda,
                const float* __restrict__ inB, int ldb,
                const float* __restrict__ W,
                const float* __restrict__ bias,
                const float* __restrict__ scale,
                float* __restrict__ out, int nrows)
{
    const int lane = threadIdx.x & 31;
    const int wid  = threadIdx.x >> 5;
    const int tile = blockIdx.x * 8 + wid;
    const int row0 = tile * 16;
    if (row0 >= nrows) return;          // wave-uniform exit (EXEC stays all-1 for WMMA)

    const int half = (lane >> 4) & 1;   // 0 or 1
    const int l    = lane & 15;

    const float* __restrict__ aRow = inA + (size_t)(row0 + l) * lda;

    v8f c0 = {};                        // N columns 0..15
    v8f c1 = {};                        // N columns 16..31

    // ---- segment A ----
    #pragma unroll
    for (int kk = 0; kk < KA; kk += 4) {
        const int ka = kk + half * 2;
        v2f a, b0, b1;
        if constexpr ((KA & 3) == 0) {
            const float2 av = *(const float2*)(aRow + ka);
            a.x = av.x; a.y = av.y;
            b0.x = W[(size_t)(ka    ) * 32 + l     ];
            b0.y = W[(size_t)(ka + 1) * 32 + l     ];
            b1.x = W[(size_t)(ka    ) * 32 + 16 + l];
            b1.y = W[(size_t)(ka + 1) * 32 + 16 + l];
        } else {
            // unconditional loads from clamped (always valid) addresses,
            // zero the VALUE with cndmask -> no exec-mask branches
            const int k0 = (ka     < KA) ? ka     : (KA - 1);
            const int k1 = (ka + 1 < KA) ? ka + 1 : (KA - 1);
            const float a0  = aRow[k0];
            const float a1  = aRow[k1];
            const float w00 = W[(size_t)k0 * 32 + l];
            const float w01 = W[(size_t)k1 * 32 + l];
            const float w10 = W[(size_t)k0 * 32 + 16 + l];
            const float w11 = W[(size_t)k1 * 32 + 16 + l];
            const bool v0 = (ka     < KA);
            const bool v1 = (ka + 1 < KA);
            a.x  = v0 ? a0  : 0.0f;  a.y  = v1 ? a1  : 0.0f;
            b0.x = v0 ? w00 : 0.0f;  b0.y = v1 ? w01 : 0.0f;
            b1.x = v0 ? w10 : 0.0f;  b1.y = v1 ? w11 : 0.0f;
        }
        c0 = __builtin_amdgcn_wmma_f32_16x16x4_f32(false, a, false, b0, (short)0, c0, false, false);
        c1 = __builtin_amdgcn_wmma_f32_16x16x4_f32(false, a, false, b1, (short)0, c1, false, false);
    }

    // ---- segment B (second half of a virtual concat; W rows offset by KA) ----
    if constexpr (KB > 0) {
        static_assert((KB & 3) == 0, "KB must be a multiple of 4");
        const float* __restrict__ bRow = inB + (size_t)(row0 + l) * ldb;
        #pragma unroll
        for (int kk = 0; kk < KB; kk += 4) {
            const int kb = kk + half * 2;
            v2f a, b0, b1;
            const float2 av = *(const float2*)(bRow + kb);
            a.x = av.x; a.y = av.y;
            b0.x = W[(size_t)(KA + kb    ) * 32 + l     ];
            b0.y = W[(size_t)(KA + kb + 1) * 32 + l     ];
            b1.x = W[(size_t)(KA + kb    ) * 32 + 16 + l];
            b1.y = W[(size_t)(KA + kb + 1) * 32 + 16 + l];
            c0 = __builtin_amdgcn_wmma_f32_16x16x4_f32(false, a, false, b0, (short)0, c0, false, false);
            c1 = __builtin_amdgcn_wmma_f32_16x16x4_f32(false, a, false, b1, (short)0, c1, false, false);
        }
    }

    #pragma unroll
    for (int v = 0; v < 8; ++v) {
        const int row = row0 + v + half * 8;
        float v0 = c0[v];
        float v1 = c1[v];
        if constexpr (BIAS)  { v0 += bias[l]; v1 += bias[16 + l]; }
        if constexpr (SCALE) { const float s = scale[row]; v0 *= s; v1 *= s; }
        out[(size_t)row * 32 + l     ] = v0;
        out[(size_t)row * 32 + 16 + l] = v1;
    }
}

// ---------------- edge scatter: acc[dst] += hw[src] (32 floats/edge, 8 thr/edge) ----------------
__global__ __launch_bounds__(256)
void scatter_edges(const int* __restrict__ src, const int* __restrict__ dst,
                   const float* __restrict__ hw, float* __restrict__ acc, int nE)
{
    int t = blockIdx.x * blockDim.x + threadIdx.x;
    int e = t >> 3;
    if (e >= nE) return;
    int c = (t & 7) * 4;
    int s = src[e];
    int d = dst[e];
    const float4 v = *(const float4*)(hw + (size_t)s * 32 + c);
    float* a = acc + (size_t)d * 32 + c;
    atomicAdd(a + 0, v.x);
    atomicAdd(a + 1, v.y);
    atomicAdd(a + 2, v.z);
    atomicAdd(a + 3, v.w);
}

// conv epilogue: h1 = relu(dis[i] * (acc + hw) + b[c])   (hw term = self loop, since hw = dis*hW)
__global__ __launch_bounds__(256)
void conv_post(const float* __restrict__ acc, const float* __restrict__ hw,
               const float* __restrict__ dis, const float* __restrict__ b,
               float* __restrict__ h1, int n)
{
    int t = blockIdx.x * blockDim.x + threadIdx.x;
    if (t >= n * 32) return;
    int i = t >> 5, c = t & 31;
    float v = dis[i] * (acc[t] + hw[t]) + b[c];
    h1[t] = fmaxf(v, 0.0f);
}

// fused conv2 epilogue + prediction head: one wave per node, lane = feature
__global__ __launch_bounds__(256)
void final_kernel(const float* __restrict__ acc2, const float* __restrict__ hw2,
                  const float* __restrict__ dis, const float* __restrict__ b2,
                  const float* __restrict__ h0, const float* __restrict__ Wp,
                  const float* __restrict__ b_pred, const float* __restrict__ x,
                  float* __restrict__ out, int n)
{
    int i = (blockIdx.x * blockDim.x + threadIdx.x) >> 5;   // wave-uniform
    int lane = threadIdx.x & 31;
    if (i >= n) return;
    size_t base = (size_t)i * 32 + lane;
    float d = dis[i];
    float h2c = fmaxf(d * (acc2[base] + hw2[base]) + b2[lane], 0.0f);
    float part = h2c * Wp[lane] + h0[base] * Wp[32 + lane];
    #pragma unroll
    for (int off = 16; off > 0; off >>= 1)
        part += __shfl_xor(part, off, 32);
    if (lane == 0) {
        float delta = part + b_pred[0] + x[(size_t)i * 22 + 1];
        out[i] = fmaxf(delta, 0.0f);
    }
}

// ---------------- launcher ----------------
extern "C" void kernel_launch(void* const* d_in, const int* in_sizes, int n_in,
                              void* d_out, int out_size, void* d_ws, size_t ws_size,
                              hipStream_t stream) {
    const float* x       = (const float*)d_in[0];
    const int*   ei      = (const int*)  d_in[1];
    const float* W_embed = (const float*)d_in[2];
    const float* b_embed = (const float*)d_in[3];
    const float* W1      = (const float*)d_in[4];
    const float* b1      = (const float*)d_in[5];
    const float* W2      = (const float*)d_in[6];
    const float* b2      = (const float*)d_in[7];
    const float* Wp      = (const float*)d_in[8];
    const float* b_pred  = (const float*)d_in[9];
    float* out = (float*)d_out;

    const int* src = ei;              // edge_index[0]
    const int* dst = ei + NE;         // edge_index[1]

    float* ws  = (float*)d_ws;
    float* dis = ws;                  // NN
    float* h0  = ws + 100352;         // NN*32
    float* A   = h0 + 3200000;        // hw1, later acc2
    float* B   = A  + 3200000;        // acc1, later hw2
    float* C   = B  + 3200000;        // h1

    const int NB_N   = (NN + 255) / 256;            // 391
    const int NB_E   = (NE + 255) / 256;            // 6250
    const int NB_G   = ((NN / 16) + 7) / 8;         // 782 blocks, 8 wave-tiles each
    const int NB_NF  = (NN * 32 + 255) / 256;       // 12500
    const int NB_SC  = (NE * 8 + 255) / 256;        // 50000

    // degrees -> dis = rsqrt(deg+1)
    zero_f32<<<NB_N, 256, 0, stream>>>(dis, (size_t)NN);
    degree_kernel<<<NB_E, 256, 0, stream>>>(dst, dis, NE);
    rsqrt_kernel<<<NB_N, 256, 0, stream>>>(dis, NN);

    // h0 = x @ W_embed + b_embed
    gemm_tiles<22, 0, true, false><<<NB_G, 256, 0, stream>>>(
        x, 22, nullptr, 0, W_embed, b_embed, nullptr, h0, NN);

    // conv1: hw1 = dis * (h0 @ W1); acc1 = scatter(hw1); h1 = relu(dis*(acc1+hw1)+b1)
    gemm_tiles<32, 0, false, true><<<NB_G, 256, 0, stream>>>(
        h0, 32, nullptr, 0, W1, nullptr, dis, A, NN);
    zero_f32<<<NB_NF, 256, 0, stream>>>(B, (size_t)NN * 32);
    scatter_edges<<<NB_SC, 256, 0, stream>>>(src, dst, A, B, NE);
    conv_post<<<NB_NF, 256, 0, stream>>>(B, A, dis, b1, C, NN);

    // conv2: hw2 = dis * (concat(h1,h0) @ W2); acc2 = scatter(hw2)
    gemm_tiles<32, 32, false, true><<<NB_G, 256, 0, stream>>>(
        C, 32, h0, 32, W2, nullptr, dis, B, NN);
    zero_f32<<<NB_NF, 256, 0, stream>>>(A, (size_t)NN * 32);
    scatter_edges<<<NB_SC, 256, 0, stream>>>(src, dst, B, A, NE);

    // fused conv2 epilogue + predict head + final relu
    final_kernel<<<NB_NF, 256, 0, stream>>>(A, B, dis, b2, h0, Wp, b_pred, x, out, NN);
}